// HebbianMambaLayer_77824807403922
// MI455X (gfx1250) — compile-verified
//
#include <hip/hip_runtime.h>
#include <hip/hip_bf16.h>
#include <stdint.h>

// ---------------------------------------------------------------------------
// Hebbian chunkwise associative memory for MI455X (gfx1250, wave32).
// Matmuls: V_WMMA_F32_16X16X4_F32 (exact fp32). Staging: TDM (tensor_load_to_lds)
// with descriptor-level LDS padding, double-buffered via TENSORcnt.
// ---------------------------------------------------------------------------

typedef __attribute__((ext_vector_type(2))) float v2f;
typedef __attribute__((ext_vector_type(8))) float v8f;
typedef __attribute__((ext_vector_type(4))) unsigned int u32x4;
typedef __attribute__((ext_vector_type(8))) int i32x8;
typedef __attribute__((ext_vector_type(4))) int i32x4;

static constexpr int Bn = 4;
static constexpr int Tn = 8192;
static constexpr int Dn = 512;
static constexpr int Cn = 64;
static constexpr int NC = Tn / Cn;      // 128 chunks
static constexpr float ALPHA = 0.03f;

__device__ __forceinline__ v8f wmma4(v2f a, v2f b, v8f c) {
  return __builtin_amdgcn_wmma_f32_16x16x4_f32(
      false, a, false, b, (short)0, c, false, false);
}

__device__ __forceinline__ unsigned lds_addr32(const void* p) {
  // generic (addrspacecast from LDS) pointer: low 32 bits are the LDS offset
  return (unsigned)(uint64_t)(uintptr_t)p;
}

// TDM 2D tile load: tile_w fp32 elements x tile_h rows, global row stride
// stride_elts, into LDS with 1 DWORD of padding after every tile_w DWORDs
// (pad_code: 3 -> every 16 DW, 4 -> every 32 DW, 5 -> every 64 DW).
__device__ __forceinline__ void tdm_load_2d(unsigned lds_byte, const float* gsrc,
                                            unsigned tile_w, unsigned tile_h,
                                            unsigned stride_elts, unsigned pad_code) {
  const uint64_t ga = (uint64_t)(uintptr_t)gsrc;
  u32x4 g0;
  g0[0] = 1u;                                              // count = 1
  g0[1] = lds_byte;                                        // LDS dest (bytes)
  g0[2] = (unsigned)ga;                                    // global_addr[31:0]
  g0[3] = (unsigned)((ga >> 32) & 0x01ffffffull) | (2u << 30);  // addr[56:32] | type=2
  i32x8 g1;
  g1[0] = (int)((2u << 16) | (1u << 20) | (pad_code << 22)); // 4B elems, pad_en, pad=1 DW
  g1[1] = (int)(tile_w << 16);                             // tensor_dim0[15:0]
  g1[2] = (int)(tile_h << 16);                             // tensor_dim1[15:0]
  g1[3] = (int)(tile_w << 16);                             // tile_dim0
  g1[4] = (int)tile_h;                                     // tile_dim1 (tile_dim2=0)
  g1[5] = (int)stride_elts;                                // tensor_dim0_stride[31:0]
  g1[6] = 0;
  g1[7] = 0;
  i32x4 z4 = {};
  i32x8 z8 = {};
  __builtin_amdgcn_tensor_load_to_lds(g0, g1, z4, z4, z8, 0);
}

template <int N>
__device__ __forceinline__ void tensor_wait() {
  __builtin_amdgcn_s_wait_tensorcnt(N);
}

// ---------------------------------------------------------------------------
// GEMM: Y = base + alpha * (X @ Wm^T).  X: Mrows x D, Wm: D x D, row-major.
// 64x64 tile / block, 8 waves * two 16x16 subtiles, TDM double-buffered K.
// ---------------------------------------------------------------------------
__global__ __launch_bounds__(256)
void gemm_xwt(const float* __restrict__ X, const float* __restrict__ Wm,
              const float* __restrict__ base, float* __restrict__ Y,
              float alpha) {
  __shared__ float sA[2][64 * 17];
  __shared__ float sB[2][64 * 17];

  const int m0 = blockIdx.y * 64;
  const int n0 = blockIdx.x * 64;
  const int tid = threadIdx.x;
  const int lane = tid & 31;
  const int wave = tid >> 5;
  const int half = lane >> 4;
  const int ln = lane & 15;
  const int tm = wave >> 1;
  const int tn = wave & 1;

  v8f acc0 = {};
  v8f acc1 = {};

  if (wave == 0) {
    tdm_load_2d(lds_addr32(sA[0]), X + (size_t)m0 * Dn, 16, 64, Dn, 3);
    tdm_load_2d(lds_addr32(sB[0]), Wm + (size_t)n0 * Dn, 16, 64, Dn, 3);
  }

  for (int k0 = 0; k0 < Dn; k0 += 16) {
    const int buf = (k0 >> 4) & 1;
    if (wave == 0) {
      if (k0 + 16 < Dn) {
        tdm_load_2d(lds_addr32(sA[buf ^ 1]), X + (size_t)m0 * Dn + k0 + 16, 16, 64, Dn, 3);
        tdm_load_2d(lds_addr32(sB[buf ^ 1]), Wm + (size_t)n0 * Dn + k0 + 16, 16, 64, Dn, 3);
        tensor_wait<2>();         // oldest pair (current buf) complete
      } else {
        tensor_wait<0>();
      }
    }
    __syncthreads();
    const float* A = sA[buf];
    const float* Bs = sB[buf];
#pragma unroll
    for (int kk = 0; kk < 16; kk += 4) {
      const int ka = kk + half * 2;
      v2f a, b0, b1;
      a.x = A[(tm * 16 + ln) * 17 + ka];
      a.y = A[(tm * 16 + ln) * 17 + ka + 1];
      b0.x = Bs[(tn * 16 + ln) * 17 + ka];
      b0.y = Bs[(tn * 16 + ln) * 17 + ka + 1];
      b1.x = Bs[((tn + 2) * 16 + ln) * 17 + ka];
      b1.y = Bs[((tn + 2) * 16 + ln) * 17 + ka + 1];
      acc0 = wmma4(a, b0, acc0);
      acc1 = wmma4(a, b1, acc1);
    }
    __syncthreads();
  }

#pragma unroll
  for (int r = 0; r < 8; ++r) {
    const int row = m0 + tm * 16 + r + half * 8;
    const size_t i0 = (size_t)row * Dn + n0 + tn * 16 + ln;
    const size_t i1 = (size_t)row * Dn + n0 + (tn + 2) * 16 + ln;
    float y0 = alpha * acc0[r];
    float y1 = alpha * acc1[r];
    if (base) { y0 += base[i0]; y1 += base[i1]; }
    Y[i0] = y0;
    Y[i1] = y1;
  }
}

// ---------------------------------------------------------------------------
// Chunkwise Hebbian scan. One block per (batch, 32-row slice of W).
// ---------------------------------------------------------------------------
static constexpr int DS = 32;
static constexpr int P_LD = 65;
static constexpr int W_LD = 516;
static constexpr int S_LD = 65;
static constexpr int V_LD = 33;
static constexpr unsigned PBYTES = 65u * (unsigned)P_LD * 4u;

// Panel = out rows [c*64-1 .. c*64+63], 64 cols starting at ep.
// Chunk 0: TDM loads 64 rows into LDS rows 1..64; row 0 zeroed by threads.
__device__ __forceinline__ void issue_panel(unsigned sPoff, int buf,
                                            const float* outb, int c, int ep) {
  const unsigned dst = sPoff + (unsigned)buf * PBYTES;
  const int row0 = c * 64 - 1;
  if (row0 >= 0) {
    tdm_load_2d(dst, outb + (size_t)row0 * Dn + ep, 64, 65, Dn, 5);
  } else {
    tdm_load_2d(dst + (unsigned)P_LD * 4u, outb + ep, 64, 64, Dn, 5);
  }
}

__global__ __launch_bounds__(256)
void hebbian_scan(const float* __restrict__ out, const float* __restrict__ v,
                  const float* __restrict__ decay, float* __restrict__ reads) {
  __shared__ float sW[DS * W_LD];          // 66.0 KB : W[d0:d0+32, 0:512]
  __shared__ float sP[2][65 * P_LD];       // 33.8 KB : double-buffered panels
  __shared__ float sS[Cn * S_LD];          // 16.6 KB : masked S
  __shared__ float sV[Cn * V_LD];          //  8.4 KB : v chunk slice

  const int b = blockIdx.x >> 4;
  const int d0 = (blockIdx.x & 15) * DS;
  const int tid = threadIdx.x;
  const int lane = tid & 31;
  const int wave = tid >> 5;
  const int half = lane >> 4;
  const int ln = lane & 15;
  const int tm = wave >> 1;   // t-tile (0..3)
  const int tn = wave & 1;    // d-tile in slice (0..1)

  const float dec = decay[0];
  const float gamma = 1.0f / (1.0f + __expf(-dec));
  const float lg = __logf(gamma);
  const float gC = __expf(64.0f * lg);

  for (int i = tid; i < DS * W_LD; i += 256) sW[i] = 0.0f;
  __syncthreads();

  const float* outb = out + (size_t)b * Tn * Dn;
  const float* vb = v + (size_t)b * Tn * Dn;
  float* rb = reads + (size_t)b * Tn * Dn;

  const unsigned sPoff = lds_addr32(&sP[0][0]);
  const unsigned sVoff = lds_addr32(&sV[0]);

  for (int c = 0; c < NC; ++c) {
    v8f accI = {};
    v8f accS0 = {};
    v8f accS1 = {};

    // ---- Phase 1: stream K-panels via TDM; accumulate inter and S --------
    if (wave == 0) {
      tdm_load_2d(sVoff, vb + (size_t)(c * 64) * Dn + d0, 32, 64, Dn, 4);
      issue_panel(sPoff, 0, outb, c, 0);
    }
    for (int epi = 0; epi < 8; ++epi) {
      const int buf = epi & 1;
      const int ep = epi * 64;
      float* sPb = sP[buf];
      if (wave == 0) {
        if (epi < 7) { issue_panel(sPoff, buf ^ 1, outb, c, ep + 64); tensor_wait<1>(); }
        else tensor_wait<0>();
      }
      if (c == 0 && tid < 64) sPb[tid] = 0.0f;   // zero wk row t=-1
      __syncthreads();
#pragma unroll
      for (int kk = 0; kk < 64; kk += 4) {
        const int ka = kk + half * 2;
        v2f a;                                   // rkc: panel rows 1..64
        a.x = sPb[(1 + tm * 16 + ln) * P_LD + ka];
        a.y = sPb[(1 + tm * 16 + ln) * P_LD + ka + 1];
        v2f bw;                                  // B[k][d] = W[d][ep+k]
        bw.x = sW[(tn * 16 + ln) * W_LD + ep + ka];
        bw.y = sW[(tn * 16 + ln) * W_LD + ep + ka + 1];
        accI = wmma4(a, bw, accI);
        v2f bs0, bs1;                            // B[k][s] = wkc[s][ep+k]
        bs0.x = sPb[((tn * 2) * 16 + ln) * P_LD + ka];
        bs0.y = sPb[((tn * 2) * 16 + ln) * P_LD + ka + 1];
        bs1.x = sPb[((tn * 2 + 1) * 16 + ln) * P_LD + ka];
        bs1.y = sPb[((tn * 2 + 1) * 16 + ln) * P_LD + ka + 1];
        accS0 = wmma4(a, bs0, accS0);
        accS1 = wmma4(a, bs1, accS1);
      }
      __syncthreads();
    }
    if (wave == 0) issue_panel(sPoff, 0, outb, c, 0);  // prefetch for phase 3

    // ---- Phase 2: mask S; intra fused onto gamma^t-scaled inter ----------
#pragma unroll
    for (int r = 0; r < 8; ++r) {
      const int t = tm * 16 + r + half * 8;
      const int s0 = (tn * 2) * 16 + ln;
      const int s1 = (tn * 2 + 1) * 16 + ln;
      const float m0 = (t > s0) ? __expf((float)(t - 1 - s0) * lg) : 0.0f;
      const float m1 = (t > s1) ? __expf((float)(t - 1 - s1) * lg) : 0.0f;
      sS[t * S_LD + s0] = accS0[r] * m0;
      sS[t * S_LD + s1] = accS1[r] * m1;
    }
    __syncthreads();
#pragma unroll
    for (int r = 0; r < 8; ++r) {
      const int t = tm * 16 + r + half * 8;
      accI[r] *= __expf((float)t * lg);
    }
#pragma unroll
    for (int kk = 0; kk < 64; kk += 4) {         // += (S.M) @ vc
      const int ka = kk + half * 2;
      v2f a, bv;
      a.x = sS[(tm * 16 + ln) * S_LD + ka];
      a.y = sS[(tm * 16 + ln) * S_LD + ka + 1];
      bv.x = sV[ka * V_LD + tn * 16 + ln];
      bv.y = sV[(ka + 1) * V_LD + tn * 16 + ln];
      accI = wmma4(a, bv, accI);
    }
#pragma unroll
    for (int r = 0; r < 8; ++r) {
      const int t = tm * 16 + r + half * 8;
      rb[(size_t)(c * 64 + t) * Dn + d0 + tn * 16 + ln] = accI[r];
    }
    __syncthreads();

    // ---- Phase 3: W = gC*W + (vc*gw)^T @ wkc, TDM-streamed panels --------
    for (int epi = 0; epi < 8; ++epi) {
      const int buf = epi & 1;
      const int ep = epi * 64;
      float* sPb = sP[buf];
      if (wave == 0) {
        if (epi < 7) { issue_panel(sPoff, buf ^ 1, outb, c, ep + 64); tensor_wait<1>(); }
        else tensor_wait<0>();
      }
      if (c == 0 && tid < 64) sPb[tid] = 0.0f;
      __syncthreads();
      const int dt = wave & 1;
      const int e0 = (wave >> 1) * 16;
      v8f cw;
#pragma unroll
      for (int r = 0; r < 8; ++r)
        cw[r] = sW[(dt * 16 + r + half * 8) * W_LD + ep + e0 + ln] * gC;
#pragma unroll
      for (int kk = 0; kk < 64; kk += 4) {
        const int ka = kk + half * 2;
        v2f a, bp;                     // A[d][t] = vc[t][d]*gw[t]
        const float gw0 = __expf((float)(63 - ka) * lg);
        const float gw1 = __expf((float)(62 - ka) * lg);
        a.x = sV[ka * V_LD + dt * 16 + ln] * gw0;
        a.y = sV[(ka + 1) * V_LD + dt * 16 + ln] * gw1;
        bp.x = sPb[ka * P_LD + e0 + ln];
        bp.y = sPb[(ka + 1) * P_LD + e0 + ln];
        cw = wmma4(a, bp, cw);
      }
#pragma unroll
      for (int r = 0; r < 8; ++r)
        sW[(dt * 16 + r + half * 8) * W_LD + ep + e0 + ln] = cw[r];
      __syncthreads();
    }
  }
}

// ---------------------------------------------------------------------------
extern "C" void kernel_launch(void* const* d_in, const int* in_sizes, int n_in,
                              void* d_out, int out_size, void* d_ws, size_t ws_size,
                              hipStream_t stream) {
  const float* out_in  = (const float*)d_in[0];   // (B,T,D) f32
  const float* w_write = (const float*)d_in[1];   // (D,D)   f32
  const float* w_read  = (const float*)d_in[2];   // (D,D)   f32
  const float* decay   = (const float*)d_in[3];   // (1,)    f32
  float* y = (float*)d_out;

  float* v_ws = (float*)d_ws;
  float* reads_ws = v_ws + (size_t)Bn * Tn * Dn;

  dim3 ggrid(Dn / 64, (Bn * Tn) / 64);

  gemm_xwt<<<ggrid, 256, 0, stream>>>(out_in, w_write, nullptr, v_ws, 1.0f);
  hebbian_scan<<<Bn * (Dn / DS), 256, 0, stream>>>(out_in, v_ws, decay, reads_ws);
  gemm_xwt<<<ggrid, 256, 0, stream>>>(reads_ws, w_read, out_in, y, ALPHA);
}